// MessageAttention_88854283420025
// MI455X (gfx1250) — compile-verified
//
#include <hip/hip_runtime.h>
#include <hip/hip_bf16.h>
#include <math.h>

typedef __attribute__((ext_vector_type(2))) float v2f;
typedef __attribute__((ext_vector_type(8))) float v8f;

// D = A(16x4) * B(4x16) + C   (CDNA5 f32 WMMA, wave32)
#define WMMA_F32_4(ACC, A, B)                                                  \
  ACC = __builtin_amdgcn_wmma_f32_16x16x4_f32(false, (A), false, (B),          \
                                              (short)0, (ACC), false, false)

// Monotonic float<->uint mapping so integer atomicMax == float max.
__device__ __forceinline__ unsigned fenc(float f) {
  unsigned u = __float_as_uint(f);
  return u ^ ((u >> 31) ? 0xFFFFFFFFu : 0x80000000u);
}
__device__ __forceinline__ float fdec(unsigned u) {
  unsigned b = u ^ ((u >> 31) ? 0x80000000u : 0xFFFFFFFFu);
  return __uint_as_float(b);
}
__device__ __forceinline__ void atomAddF(float* p, float v) {
  unsafeAtomicAdd(p, v);  // hardware global_atomic_add_f32
}

// ---------------------------------------------------------------------------
// GDB kernel: per 16-node tile (one wave each), all GEMMs via WMMA f32 16x16x4.
// LDS: weights (18464 f) + per-wave scratch 4608 f (vi:1536, vn|z:1024, out_s:2048)
// ---------------------------------------------------------------------------
__global__ __launch_bounds__(128) void gdb_kernel(
    const float* __restrict__ x_sca, const float* __restrict__ x_vec,
    const float* __restrict__ Wv1, const float* __restrict__ Wv2,
    const float* __restrict__ Ws1, const float* __restrict__ Ws2,
    const float* __restrict__ Wg, const float* __restrict__ bg,
    float* __restrict__ outS, float* __restrict__ outV, int N) {
  extern __shared__ float smem[];
  float* sWv1 = smem;          // 1024  (32x32, [h][c])
  float* sWv2 = smem + 1024;   // 1024  (32x32, [o][h])
  float* sWs1 = smem + 2048;   // 4096  (32x128, [h][c])
  float* sWs2 = smem + 6144;   // 8192  (128x64, [o][k])
  float* sWg  = smem + 14336;  // 4096  (32x128, [o][k])
  float* sBg  = smem + 18432;  // 32

  const int tid = threadIdx.x;
  for (int i = tid; i < 1024; i += 128) sWv1[i] = Wv1[i];
  for (int i = tid; i < 1024; i += 128) sWv2[i] = Wv2[i];
  for (int i = tid; i < 4096; i += 128) sWs1[i] = Ws1[i];
  for (int i = tid; i < 8192; i += 128) sWs2[i] = Ws2[i];
  for (int i = tid; i < 4096; i += 128) sWg[i] = Wg[i];
  if (tid < 32) sBg[tid] = bg[tid];
  __syncthreads();

  const int wave = tid >> 5, lane = tid & 31;
  const int lr = lane & 15, hl = lane >> 4;  // hl selects K-pair / row-half
  float* sVi = smem + 18464 + wave * 4608;   // [3][16 nodes][32 hid]
  float* sT  = sVi + 1536;                   // [16][64] vn|z ; +1024: [16][128] out_s
  const int n0 = (blockIdx.x * 4 + wave) * 16;
  const int nA = min(n0 + lr, N - 1);  // clamped node for A-fragment loads

  // ---- vi[i] = x_vec[:, :, i] @ Wv1^T  (16x32 = 16x32 * 32x32), per axis ----
  v8f vi_acc[3][2];
  for (int i = 0; i < 3; ++i)
    for (int ct = 0; ct < 2; ++ct) {
      v8f acc = {0, 0, 0, 0, 0, 0, 0, 0};
#pragma unroll
      for (int kc = 0; kc < 8; ++kc) {
        const int k0 = kc * 4 + hl * 2;
        v2f a, b;
        a.x = x_vec[(size_t)nA * 96 + (k0 + 0) * 3 + i];
        a.y = x_vec[(size_t)nA * 96 + (k0 + 1) * 3 + i];
        b.x = sWv1[(lr + 16 * ct) * 32 + k0];
        b.y = sWv1[(lr + 16 * ct) * 32 + k0 + 1];
        WMMA_F32_4(acc, a, b);
      }
      vi_acc[i][ct] = acc;
    }
  // spill vi to LDS (A-layout source for later GEMM) and build vn into sT
  for (int ct = 0; ct < 2; ++ct)
#pragma unroll
    for (int r = 0; r < 8; ++r) {
      const int row = r + 8 * hl, col = lr + 16 * ct;
      const float vx = vi_acc[0][ct][r], vy = vi_acc[1][ct][r], vz = vi_acc[2][ct][r];
      sVi[0 * 512 + row * 32 + col] = vx;
      sVi[1 * 512 + row * 32 + col] = vy;
      sVi[2 * 512 + row * 32 + col] = vz;
      sT[row * 64 + col] = sqrtf(vx * vx + vy * vy + vz * vz);
    }

  // ---- z = x_sca @ Ws1^T   (16x32 = 16x128 * 128x32) ----
  for (int ct = 0; ct < 2; ++ct) {
    v8f acc = {0, 0, 0, 0, 0, 0, 0, 0};
#pragma unroll
    for (int kc = 0; kc < 32; ++kc) {
      const int k0 = kc * 4 + hl * 2;
      v2f a, b;
      a.x = x_sca[(size_t)nA * 128 + k0];
      a.y = x_sca[(size_t)nA * 128 + k0 + 1];
      b.x = sWs1[(lr + 16 * ct) * 128 + k0];
      b.y = sWs1[(lr + 16 * ct) * 128 + k0 + 1];
      WMMA_F32_4(acc, a, b);
    }
#pragma unroll
    for (int r = 0; r < 8; ++r)
      sT[(r + 8 * hl) * 64 + 32 + lr + 16 * ct] = acc[r];
  }

  // ---- out_s = [vn | z] @ Ws2^T   (16x128 = 16x64 * 64x128) ----
  for (int ct = 0; ct < 8; ++ct) {
    v8f acc = {0, 0, 0, 0, 0, 0, 0, 0};
#pragma unroll
    for (int kc = 0; kc < 16; ++kc) {
      const int k0 = kc * 4 + hl * 2;
      v2f a, b;
      a.x = sT[lr * 64 + k0];
      a.y = sT[lr * 64 + k0 + 1];
      b.x = sWs2[(lr + 16 * ct) * 64 + k0];
      b.y = sWs2[(lr + 16 * ct) * 64 + k0 + 1];
      WMMA_F32_4(acc, a, b);
    }
#pragma unroll
    for (int r = 0; r < 8; ++r) {
      const int row = r + 8 * hl, col = lr + 16 * ct;
      sT[1024 + row * 128 + col] = acc[r];
      if (n0 + row < N) outS[(size_t)(n0 + row) * 128 + col] = acc[r];
    }
  }

  // ---- gate = sigmoid(out_s @ Wg^T + bg)   (16x32) ----
  v8f gate[2];
  for (int ct = 0; ct < 2; ++ct) {
    v8f acc = {0, 0, 0, 0, 0, 0, 0, 0};
#pragma unroll
    for (int kc = 0; kc < 32; ++kc) {
      const int k0 = kc * 4 + hl * 2;
      v2f a, b;
      a.x = sT[1024 + lr * 128 + k0];
      a.y = sT[1024 + lr * 128 + k0 + 1];
      b.x = sWg[(lr + 16 * ct) * 128 + k0];
      b.y = sWg[(lr + 16 * ct) * 128 + k0 + 1];
      WMMA_F32_4(acc, a, b);
    }
    const float bias = sBg[lr + 16 * ct];
#pragma unroll
    for (int r = 0; r < 8; ++r)
      acc[r] = 1.f / (1.f + __expf(-(acc[r] + bias)));
    gate[ct] = acc;
  }

  // ---- out_v = gate * (vi @ Wv2^T)   (per axis, 16x32 = 16x32 * 32x32) ----
  for (int i = 0; i < 3; ++i)
    for (int ct = 0; ct < 2; ++ct) {
      v8f acc = {0, 0, 0, 0, 0, 0, 0, 0};
#pragma unroll
      for (int kc = 0; kc < 8; ++kc) {
        const int k0 = kc * 4 + hl * 2;
        v2f a, b;
        a.x = sVi[i * 512 + lr * 32 + k0];
        a.y = sVi[i * 512 + lr * 32 + k0 + 1];
        b.x = sWv2[(lr + 16 * ct) * 32 + k0];
        b.y = sWv2[(lr + 16 * ct) * 32 + k0 + 1];
        WMMA_F32_4(acc, a, b);
      }
#pragma unroll
      for (int r = 0; r < 8; ++r) {
        const int row = r + 8 * hl, col = lr + 16 * ct;
        if (n0 + row < N)
          outV[(size_t)(n0 + row) * 96 + col * 3 + i] = gate[ct][r] * acc[r];
      }
    }
}

// ---------------------------------------------------------------------------
// Edge-side kernels (bandwidth bound; atomics-based segment softmax)
// ---------------------------------------------------------------------------
__global__ void init_kernel(unsigned* mx_s, unsigned* mx_v, float* den_s,
                            float* den_v, int n) {
  int i = blockIdx.x * blockDim.x + threadIdx.x;
  if (i < n) {
    mx_s[i] = 0x007FFFFFu;  // fenc(-inf)
    mx_v[i] = 0x007FFFFFu;
    den_s[i] = 0.f;
    den_v[i] = 0.f;
  }
}

__global__ void logits_kernel(const float* __restrict__ q_sca,
                              const float* __restrict__ q_vec,
                              const float* __restrict__ k_s,
                              const float* __restrict__ k_v,
                              const int* __restrict__ idx, float* __restrict__ a_s,
                              float* __restrict__ a_v, unsigned* mx_s,
                              unsigned* mx_v, int M) {
  int m = blockIdx.x * blockDim.x + threadIdx.x;
  if (m >= M) return;
  const int e = idx[m];
  const float4* qs = reinterpret_cast<const float4*>(q_sca) + (size_t)m * 32;
  const float4* ks = reinterpret_cast<const float4*>(k_s) + (size_t)e * 32;
  const float4* qv = reinterpret_cast<const float4*>(q_vec) + (size_t)m * 24;
  const float4* kv = reinterpret_cast<const float4*>(k_v) + (size_t)e * 24;
#pragma unroll
  for (int h = 0; h < 4; ++h) {
    float s = 0.f;
#pragma unroll
    for (int j = 0; j < 8; ++j) {
      float4 a = qs[h * 8 + j], b = ks[h * 8 + j];
      s += a.x * b.x + a.y * b.y + a.z * b.z + a.w * b.w;
    }
    a_s[(size_t)m * 4 + h] = s;
    atomicMax(&mx_s[(size_t)e * 4 + h], fenc(s));
    float v = 0.f;
#pragma unroll
    for (int j = 0; j < 6; ++j) {
      float4 a = qv[h * 6 + j], b = kv[h * 6 + j];
      v += a.x * b.x + a.y * b.y + a.z * b.z + a.w * b.w;
    }
    a_v[(size_t)m * 4 + h] = v;
    atomicMax(&mx_v[(size_t)e * 4 + h], fenc(v));
  }
}

__global__ void softexp_kernel(float* __restrict__ a_s, float* __restrict__ a_v,
                               const int* __restrict__ idx,
                               const unsigned* __restrict__ mx_s,
                               const unsigned* __restrict__ mx_v, float* den_s,
                               float* den_v, int M) {
  int m = blockIdx.x * blockDim.x + threadIdx.x;
  if (m >= M) return;
  const int e = idx[m];
#pragma unroll
  for (int h = 0; h < 4; ++h) {
    float es = __expf(a_s[(size_t)m * 4 + h] - fdec(mx_s[(size_t)e * 4 + h]));
    a_s[(size_t)m * 4 + h] = es;  // overwrite logits with exp in place
    atomAddF(&den_s[(size_t)e * 4 + h], es);
    float ev = __expf(a_v[(size_t)m * 4 + h] - fdec(mx_v[(size_t)e * 4 + h]));
    a_v[(size_t)m * 4 + h] = ev;
    atomAddF(&den_v[(size_t)e * 4 + h], ev);
  }
}

__global__ void scatter_kernel(const float* __restrict__ q_sca,
                               const float* __restrict__ q_vec,
                               const int* __restrict__ idx,
                               const float* __restrict__ a_s,
                               const float* __restrict__ a_v,
                               const float* __restrict__ den_s,
                               const float* __restrict__ den_v,
                               float* outS, float* outV, int M) {
  int m = blockIdx.x * blockDim.x + threadIdx.x;
  if (m >= M) return;
  const int e = idx[m];
#pragma unroll
  for (int h = 0; h < 4; ++h) {
    const float wS = a_s[(size_t)m * 4 + h] / den_s[(size_t)e * 4 + h];
#pragma unroll
    for (int j = 0; j < 32; ++j)
      atomAddF(&outS[(size_t)e * 128 + h * 32 + j],
               wS * q_sca[(size_t)m * 128 + h * 32 + j]);
    const float wV = a_v[(size_t)m * 4 + h] / den_v[(size_t)e * 4 + h];
#pragma unroll
    for (int j = 0; j < 24; ++j)
      atomAddF(&outV[(size_t)e * 96 + h * 24 + j],
               wV * q_vec[(size_t)m * 96 + h * 24 + j]);
  }
}

// ---------------------------------------------------------------------------
extern "C" void kernel_launch(void* const* d_in, const int* in_sizes, int n_in,
                              void* d_out, int out_size, void* d_ws,
                              size_t ws_size, hipStream_t stream) {
  (void)n_in; (void)out_size; (void)ws_size;
  const float* x_sca = (const float*)d_in[0];
  const float* x_vec = (const float*)d_in[1];
  const float* q_sca = (const float*)d_in[2];
  const float* q_vec = (const float*)d_in[3];
  const int* eidx = (const int*)d_in[4];
  const float* lkWv1 = (const float*)d_in[5];
  const float* lkWv2 = (const float*)d_in[6];
  const float* lkWs1 = (const float*)d_in[7];
  const float* lkWs2 = (const float*)d_in[8];
  const float* lkWg = (const float*)d_in[9];
  const float* lkBg = (const float*)d_in[10];
  const float* lvWv1 = (const float*)d_in[11];
  const float* lvWv2 = (const float*)d_in[12];
  const float* lvWs1 = (const float*)d_in[13];
  const float* lvWs2 = (const float*)d_in[14];
  const float* lvWg = (const float*)d_in[15];
  const float* lvBg = (const float*)d_in[16];

  const int N = in_sizes[0] / 128;
  const int M = in_sizes[4];

  float* outS = (float*)d_out;
  float* outV = outS + (size_t)N * 128;

  float* ws = (float*)d_ws;
  float* k_s = ws;                               // N*128
  float* k_v = k_s + (size_t)N * 128;            // N*96
  float* a_s = k_v + (size_t)N * 96;             // M*4
  float* a_v = a_s + (size_t)M * 4;              // M*4
  unsigned* mx_s = (unsigned*)(a_v + (size_t)M * 4);  // N*4
  unsigned* mx_v = mx_s + (size_t)N * 4;              // N*4
  float* den_s = (float*)(mx_v + (size_t)N * 4);      // N*4
  float* den_v = den_s + (size_t)N * 4;               // N*4

  const size_t ldsBytes = (size_t)(18464 + 4 * 4608) * sizeof(float);
  const int gdbBlocks = (N + 63) / 64;

  // k = GDB_lk(x) into workspace; root = GDB_lv(x) written straight into d_out
  gdb_kernel<<<gdbBlocks, 128, ldsBytes, stream>>>(
      x_sca, x_vec, lkWv1, lkWv2, lkWs1, lkWs2, lkWg, lkBg, k_s, k_v, N);
  gdb_kernel<<<gdbBlocks, 128, ldsBytes, stream>>>(
      x_sca, x_vec, lvWv1, lvWv2, lvWs1, lvWs2, lvWg, lvBg, outS, outV, N);

  init_kernel<<<(N * 4 + 255) / 256, 256, 0, stream>>>(mx_s, mx_v, den_s, den_v,
                                                       N * 4);
  logits_kernel<<<(M + 255) / 256, 256, 0, stream>>>(q_sca, q_vec, k_s, k_v,
                                                     eidx, a_s, a_v, mx_s, mx_v, M);
  softexp_kernel<<<(M + 255) / 256, 256, 0, stream>>>(a_s, a_v, eidx, mx_s,
                                                      mx_v, den_s, den_v, M);
  scatter_kernel<<<(M + 255) / 256, 256, 0, stream>>>(q_sca, q_vec, eidx, a_s,
                                                      a_v, den_s, den_v, outS,
                                                      outV, M);
}